// attention_68341519614767
// MI455X (gfx1250) — compile-verified
//
#include <hip/hip_runtime.h>

#define NTOK 20736   // (S*P)^2 tokens per batch = 144*144
#define DQKV 144     // input_dim
#define D3   48
#define BATCH 8

typedef float v2f __attribute__((ext_vector_type(2)));
typedef float v8f __attribute__((ext_vector_type(8)));

__device__ __forceinline__ v8f wmma_f32(v2f a, v2f b, v8f c) {
    // D = A(16x4) * B(4x16) + C(16x16), fp32 matrix pipe
    return __builtin_amdgcn_wmma_f32_16x16x4_f32(false, a, false, b, (short)0, c,
                                                 false, false);
}

// ----- gfx1250 async global->LDS staging (guarded; falls back to plain) -----
#if defined(__has_builtin)
#if __has_builtin(__builtin_amdgcn_global_load_async_to_lds_b32) && \
    __has_builtin(__builtin_amdgcn_s_wait_asynccnt)
#define USE_ASYNC_LDS 1
#endif
#endif
#ifndef USE_ASYNC_LDS
#define USE_ASYNC_LDS 0
#endif

#if USE_ASYNC_LDS
typedef __attribute__((address_space(1))) int gbl_i;
typedef __attribute__((address_space(3))) int lds_i;
__device__ __forceinline__ void stage_b32(const float* g, float* l) {
    __builtin_amdgcn_global_load_async_to_lds_b32((gbl_i*)g, (lds_i*)l, 0, 0);
}
__device__ __forceinline__ void stage_wait() {
    __builtin_amdgcn_s_wait_asynccnt(0);
}
#else
__device__ __forceinline__ void stage_b32(const float* g, float* l) { *l = *g; }
__device__ __forceinline__ void stage_wait() {}
#endif

// ---------------------------------------------------------------------------
// Kernel 1: QKV GEMM.  x[b, n=p*144+c, m=s1*12+s2] = input[b,c,s1*12+p1,s2*12+p2]
// qkv = x @ Wqkv^T + b.  Epilogue routes features: [0,48)->Q token-major,
// [48,96)->K^T image-major, [96,144)->V image-major.
// Block: 256 thr (8 waves), tile = 128 tokens x 144 feats, K staged in LDS
// chunks of 36 (async global->LDS when available).
// ---------------------------------------------------------------------------
__global__ __launch_bounds__(256) void qkv_gemm_kernel(
    const float* __restrict__ input, const float* __restrict__ Wqkv,
    const float* __restrict__ bqkv,
    float* __restrict__ Q, float* __restrict__ KT, float* __restrict__ Vimg)
{
    __shared__ float xs[128][37];   // 37 odd -> conflict-free stride
    __shared__ float wsh[144][37];

    const int b    = blockIdx.y;
    const int tok0 = blockIdx.x * 128;
    const int tid  = threadIdx.x;
    const int wave = tid >> 5;
    const int lane = tid & 31;
    const int lm   = lane & 15;
    const int hl   = lane >> 4;

    v8f acc[9];
#pragma unroll
    for (int i = 0; i < 9; ++i)
#pragma unroll
        for (int j = 0; j < 8; ++j) acc[i][j] = 0.0f;

    for (int kc = 0; kc < DQKV; kc += 36) {
        // Stage X tile (gather through the reshape/transpose chain)
        for (int e = tid; e < 128 * 36; e += 256) {
            int tm = e / 36, kk = e % 36;
            int n  = tok0 + tm;
            int c  = n % 144, p = n / 144;
            int p1 = p / 12, p2 = p % 12;
            int k  = kc + kk;
            int s1 = k / 12, s2 = k % 12;
            stage_b32(&input[(((size_t)b * 144 + c) * 144 + (s1 * 12 + p1)) * 144 +
                             (s2 * 12 + p2)],
                      &xs[tm][kk]);
        }
        // Stage W panel: wsh[f][kk] = Wqkv[f, kc+kk]
        for (int e = tid; e < 144 * 36; e += 256) {
            int f = e / 36, kk = e % 36;
            stage_b32(&Wqkv[f * 144 + kc + kk], &wsh[f][kk]);
        }
        stage_wait();
        __syncthreads();

        for (int ks = 0; ks < 36; ks += 4) {
            v2f a;
            a.x = xs[wave * 16 + lm][ks + 2 * hl + 0];
            a.y = xs[wave * 16 + lm][ks + 2 * hl + 1];
#pragma unroll
            for (int nt = 0; nt < 9; ++nt) {
                v2f bb;
                bb.x = wsh[nt * 16 + lm][ks + 2 * hl + 0];
                bb.y = wsh[nt * 16 + lm][ks + 2 * hl + 1];
                acc[nt] = wmma_f32(a, bb, acc[nt]);
            }
        }
        __syncthreads();
    }

    // Epilogue: bias + scatter to Q / K^T / V
#pragma unroll
    for (int nt = 0; nt < 9; ++nt) {
        int f = nt * 16 + lm;
        float bv = bqkv[f];
#pragma unroll
        for (int r = 0; r < 8; ++r) {
            int tok  = tok0 + wave * 16 + r + 8 * hl;
            float v  = acc[nt][r] + bv;
            if (f < 48) {
                Q[((size_t)b * NTOK + tok) * 48 + f] = v;
            } else if (f < 96) {
                KT[((size_t)b * 48 + (f - 48)) * NTOK + tok] = v;
            } else {
                Vimg[((size_t)b * 48 + (f - 96)) * NTOK + tok] = v;
            }
        }
    }
}

// ---------------------------------------------------------------------------
// Kernel 2: deterministic column sums of Q and K (for mean-pooled routing).
// grid = (48 features, 8 batches, 2 {Q,K}), block 256.
// ---------------------------------------------------------------------------
__global__ __launch_bounds__(256) void qk_sum_kernel(
    const float* __restrict__ Q, const float* __restrict__ KT,
    float* __restrict__ sums)
{
    __shared__ float red[256];
    const int d = blockIdx.x, b = blockIdx.y, z = blockIdx.z;
    float s = 0.0f;
    if (z == 0) {
        for (int n = threadIdx.x; n < NTOK; n += 256)
            s += Q[((size_t)b * NTOK + n) * 48 + d];
    } else {
        for (int n = threadIdx.x; n < NTOK; n += 256)
            s += KT[((size_t)b * 48 + d) * NTOK + n];
    }
    red[threadIdx.x] = s;
    __syncthreads();
    for (int off = 128; off > 0; off >>= 1) {
        if ((int)threadIdx.x < off) red[threadIdx.x] += red[threadIdx.x + off];
        __syncthreads();
    }
    if (threadIdx.x == 0) sums[z * 384 + b * 48 + d] = red[0];
}

// ---------------------------------------------------------------------------
// Kernel 3: routing.  A_r = q_r @ k_r^T (8x8), stable top-4 (matches lax.top_k
// tie-break), gather key_g/val_g from token rows 0..7.  One 64-thread block.
// ---------------------------------------------------------------------------
__global__ __launch_bounds__(64) void routing_kernel(
    const float* __restrict__ sums, const float* __restrict__ KT,
    const float* __restrict__ Vimg,
    float* __restrict__ keyg, float* __restrict__ valg)
{
    __shared__ float Ar[8][8];
    __shared__ int idx[8][4];
    const int t = threadIdx.x;
    const float invN = 1.0f / (float)NTOK;

    {
        int bq = t / 8, bk = t % 8;
        float s = 0.0f;
        for (int d = 0; d < 48; ++d)
            s += sums[bq * 48 + d] * sums[384 + bk * 48 + d];
        Ar[bq][bk] = s * invN * invN;
    }
    __syncthreads();
    if (t < 8) {
        float row[8];
#pragma unroll
        for (int i = 0; i < 8; ++i) row[i] = Ar[t][i];
#pragma unroll
        for (int sel = 0; sel < 4; ++sel) {
            int best = 0;
            float bv = row[0];
#pragma unroll
            for (int i = 1; i < 8; ++i)
                if (row[i] > bv) { bv = row[i]; best = i; }
            idx[t][sel] = best;
            row[best] = -3.4e38f;
        }
    }
    __syncthreads();
    for (int e = t; e < 8 * 4 * 48; e += 64) {
        int b = e / 192, r = e % 192, ts = r / 48, d = r % 48;
        int tok = idx[b][ts];              // token index in [0,8)
        keyg[e] = KT[((size_t)b * 48 + d) * NTOK + tok];
        valg[e] = Vimg[((size_t)b * 48 + d) * NTOK + tok];
    }
}

// ---------------------------------------------------------------------------
// Kernel 4: attention.  out[b,n,:] = softmax(q[b,n] . key_g[b,t]) @ val_g.
// Writes image-layout OutImg[b][d][n].  grid = (81, 8), block 256.
// ---------------------------------------------------------------------------
__global__ __launch_bounds__(256) void attn_kernel(
    const float* __restrict__ Q, const float* __restrict__ keyg,
    const float* __restrict__ valg, float* __restrict__ OutImg)
{
    __shared__ float kg[4][48];
    __shared__ float vg[4][48];
    const int b = blockIdx.y;
    const int n = blockIdx.x * 256 + threadIdx.x;

    for (int e = threadIdx.x; e < 192; e += 256) {
        kg[e / 48][e % 48] = keyg[b * 192 + e];
        vg[e / 48][e % 48] = valg[b * 192 + e];
    }
    __syncthreads();

    const float4* qr = (const float4*)(Q + ((size_t)b * NTOK + n) * 48);
    float q[48];
#pragma unroll
    for (int i = 0; i < 12; ++i) {
        float4 v = qr[i];
        q[4 * i + 0] = v.x; q[4 * i + 1] = v.y;
        q[4 * i + 2] = v.z; q[4 * i + 3] = v.w;
    }

    float s[4];
#pragma unroll
    for (int t = 0; t < 4; ++t) {
        float acc = 0.0f;
#pragma unroll
        for (int d = 0; d < 48; ++d) acc += q[d] * kg[t][d];
        s[t] = acc;
    }
    float m = fmaxf(fmaxf(s[0], s[1]), fmaxf(s[2], s[3]));
    float p[4], ssum = 0.0f;
#pragma unroll
    for (int t = 0; t < 4; ++t) { p[t] = expf(s[t] - m); ssum += p[t]; }
    float inv = 1.0f / ssum;
#pragma unroll
    for (int t = 0; t < 4; ++t) p[t] *= inv;

    for (int d = 0; d < 48; ++d) {
        float o = p[0] * vg[0][d] + p[1] * vg[1][d] + p[2] * vg[2][d] +
                  p[3] * vg[3][d];
        OutImg[((size_t)b * 48 + d) * NTOK + n] = o;
    }
}

// ---------------------------------------------------------------------------
// Kernels 5/6: 3x3 SAME conv, 48->48 ch, as implicit GEMM with WMMA f32.
// M = 16 pixels (x-contiguous), N = 48 out-ch, K = 48*9 = 432 in 4 chunks of
// 108 (12 input channels each).  Per chunk we stage both the weight panel and
// the zero-padded input slab (12 ic x 3 rows x 146 halo'd px) in LDS, so the
// inner loop is pure ds_load + v_wmma (no bounds checks, no scalar gathers).
// Block = 288 thr (9 waves) covers one full row y of one batch.
// accumulate=1: out += conv (down path on top of attention); else out = conv.
// ---------------------------------------------------------------------------
__global__ __launch_bounds__(288) void conv3x3_kernel(
    const float* __restrict__ in, const float* __restrict__ W,
    const float* __restrict__ bias, float* __restrict__ out, int accumulate)
{
    __shared__ float wsh[48][109];      // 109 odd -> conflict-free
    __shared__ float ins[12][3][146];   // [ic_local][dy][x+1], zero-padded
    const int b    = blockIdx.y;
    const int y    = blockIdx.x;
    const int tid  = threadIdx.x;
    const int wave = tid >> 5;
    const int lane = tid & 31;
    const int lm   = lane & 15;
    const int hl   = lane >> 4;
    const int x0   = wave * 16;

    v8f acc[3];
#pragma unroll
    for (int i = 0; i < 3; ++i)
#pragma unroll
        for (int j = 0; j < 8; ++j) acc[i][j] = 0.0f;

    for (int kc = 0; kc < 432; kc += 108) {
        // Weight panel: wsh[oc][kk] = W[oc*432 + kc + kk]
        for (int e = tid; e < 48 * 108; e += 288) {
            int f = e / 108, kk = e % 108;
            stage_b32(&W[f * 432 + kc + kk], &wsh[f][kk]);
        }
        // Input slab for this chunk's 12 input channels, rows y-1..y+1,
        // x = -1..144, zero-padded (cannot use async here: needs zero fill).
        for (int e = tid; e < 12 * 3 * 146; e += 288) {
            int ic_l = e / 438;
            int r    = e % 438;
            int row  = r / 146;
            int xi   = r % 146;
            int xx   = xi - 1;
            int yy   = y + row - 1;
            int ic   = kc / 9 + ic_l;
            float v  = 0.0f;
            if (yy >= 0 && yy < 144 && xx >= 0 && xx < 144)
                v = in[(((size_t)b * 48 + ic) * 144 + yy) * 144 + xx];
            ins[ic_l][row][xi] = v;
        }
        stage_wait();
        __syncthreads();

        for (int ks = 0; ks < 108; ks += 4) {
            v2f a;
#pragma unroll
            for (int j = 0; j < 2; ++j) {
                int KL   = ks + 2 * hl + j;   // local K, 0..107
                int ic_l = KL / 9, r = KL % 9;
                int dy   = r / 3, dx = r % 3;
                a[j] = ins[ic_l][dy][x0 + lm + dx];
            }
#pragma unroll
            for (int nt = 0; nt < 3; ++nt) {
                v2f bb;
                bb.x = wsh[nt * 16 + lm][ks + 2 * hl + 0];
                bb.y = wsh[nt * 16 + lm][ks + 2 * hl + 1];
                acc[nt] = wmma_f32(a, bb, acc[nt]);
            }
        }
        __syncthreads();
    }

#pragma unroll
    for (int nt = 0; nt < 3; ++nt) {
        int oc   = nt * 16 + lm;
        float bv = bias[oc];
#pragma unroll
        for (int r = 0; r < 8; ++r) {
            int x    = x0 + r + 8 * hl;
            size_t o = (((size_t)b * 48 + oc) * 144 + y) * 144 + x;
            float v  = acc[nt][r] + bv;
            out[o]   = accumulate ? (out[o] + v) : v;
        }
    }
}

// ---------------------------------------------------------------------------
extern "C" void kernel_launch(void* const* d_in, const int* in_sizes, int n_in,
                              void* d_out, int out_size, void* d_ws,
                              size_t ws_size, hipStream_t stream)
{
    const float* input = (const float*)d_in[0];
    const float* Wqkv  = (const float*)d_in[1];
    const float* bqkv  = (const float*)d_in[2];
    const float* Wdown = (const float*)d_in[3];
    const float* bdown = (const float*)d_in[4];
    const float* Wup   = (const float*)d_in[5];
    const float* bup   = (const float*)d_in[6];
    float* out = (float*)d_out;

    const size_t IMG = (size_t)BATCH * D3 * NTOK;  // 7,962,624 floats
    float* ws     = (float*)d_ws;
    float* Q      = ws;             // [B][N][48] token-major
    float* KT     = Q + IMG;        // [B][48][N] image-major
    float* Vimg   = KT + IMG;       // [B][48][N] image-major
    float* OutImg = Vimg + IMG;     // [B][48][N] image-major
    float* sums   = OutImg + IMG;   // [2][8][48]
    float* keyg   = sums + 768;     // [8][4][48]
    float* valg   = keyg + 1536;    // [8][4][48]

    qkv_gemm_kernel<<<dim3(NTOK / 128, BATCH), 256, 0, stream>>>(
        input, Wqkv, bqkv, Q, KT, Vimg);
    qk_sum_kernel<<<dim3(48, BATCH, 2), 256, 0, stream>>>(Q, KT, sums);
    routing_kernel<<<1, 64, 0, stream>>>(sums, KT, Vimg, keyg, valg);
    attn_kernel<<<dim3(NTOK / 256, BATCH), 256, 0, stream>>>(Q, keyg, valg,
                                                             OutImg);
    conv3x3_kernel<<<dim3(144, BATCH), 288, 0, stream>>>(Vimg, Wdown, bdown,
                                                         OutImg, 1);
    conv3x3_kernel<<<dim3(144, BATCH), 288, 0, stream>>>(OutImg, Wup, bup,
                                                         out, 0);
}